// GraphCDA_40553081209091
// MI455X (gfx1250) — compile-verified
//
#include <hip/hip_runtime.h>

typedef __attribute__((ext_vector_type(2))) float v2f;
typedef __attribute__((ext_vector_type(8))) float v8f;

#define D128 128
#define H8 8

// ============ WMMA f32 GEMM: C[M,N] = A[M,K] @ Bt[N,K]^T (+bias[col]) ============
// Bt is B TRANSPOSED, row-major [N,K] -> each lane's K-pair is contiguous (b64 loads).
// One wave computes a 16(M) x 32(N) tile: one A fragment, two B fragments, two
// accumulators, V_WMMA_F32_16X16X4_F32 per ISA 7.12.2 layouts:
//   A frag: lanes 0-15 rows M with K={k0,k0+1}; lanes 16-31 K={k0+2,k0+3}
//   B frag: same with N in lanes.  C/D: VGPR v, lane l -> (M=v+8*(l>=16), N=l&15).
// OOB rows/cols are CLAMPED, not zeroed: garbage lands only in C entries we never store.
__global__ void k_gemm_wmma(const float* __restrict__ A, int lda,
                            const float* __restrict__ Bt, int ldb,
                            float* __restrict__ C, int ldc,
                            const float* __restrict__ bias,
                            int M, int N, int K)
{
    const int tn   = blockIdx.x << 5;   // 32-wide N tile
    const int tm   = blockIdx.y << 4;   // 16-tall M tile
    const int lane = threadIdx.x;       // 0..31
    const int g    = lane >> 4;         // K-pair group
    const int l    = lane & 15;

    const int arow  = min(tm + l, M - 1);
    const int bcol0 = min(tn + l, N - 1);
    const int bcol1 = min(tn + 16 + l, N - 1);
    const v2f* Ap  = (const v2f*)(A  + (size_t)arow  * lda);
    const v2f* Bp0 = (const v2f*)(Bt + (size_t)bcol0 * ldb);
    const v2f* Bp1 = (const v2f*)(Bt + (size_t)bcol1 * ldb);

    v8f c0 = {}, c1 = {};
#pragma unroll 4
    for (int k0 = 0; k0 < K; k0 += 4) {
        const int kp = (k0 >> 1) + g;   // selects elements K = k0+2g, k0+2g+1
        v2f a  = Ap[kp];
        v2f b0 = Bp0[kp];
        v2f b1 = Bp1[kp];
        c0 = __builtin_amdgcn_wmma_f32_16x16x4_f32(false, a, false, b0, (short)0, c0, false, false);
        c1 = __builtin_amdgcn_wmma_f32_16x16x4_f32(false, a, false, b1, (short)0, c1, false, false);
    }

    const int col0 = tn + l, col1 = tn + 16 + l;
    const float bb0 = (bias && col0 < N) ? bias[col0] : 0.0f;
    const float bb1 = (bias && col1 < N) ? bias[col1] : 0.0f;
#pragma unroll
    for (int v = 0; v < 8; ++v) {
        const int row = tm + v + (g << 3);
        if (row < M) {
            if (col0 < N) C[(size_t)row * ldc + col0] = c0[v] + bb0;
            if (col1 < N) C[(size_t)row * ldc + col1] = c1[v] + bb1;
        }
    }
}

// ================= small utility kernels =================
__global__ void k_fillf(float* p, float v, int n) {
    int i = blockIdx.x * blockDim.x + threadIdx.x;
    if (i < n) p[i] = v;
}
__global__ void k_fillu(unsigned* p, unsigned v, int n) {
    int i = blockIdx.x * blockDim.x + threadIdx.x;
    if (i < n) p[i] = v;
}

// ew[e] = mat[src,dst]; deg[dst] += ew  (deg pre-initialized to 1.0 = self-loop weight)
__global__ void k_edge_w_deg(const int* __restrict__ src, const int* __restrict__ dst,
                             const float* __restrict__ mat, int Nn,
                             float* __restrict__ ew, float* __restrict__ deg, int E)
{
    int e = blockIdx.x * blockDim.x + threadIdx.x;
    if (e >= E) return;
    int s = src[e], d = dst[e];
    float w = mat[(size_t)s * Nn + d];
    ew[e] = w;
    atomicAdd(&deg[d], w);
}

__global__ void k_dinv(const float* __restrict__ deg, float* __restrict__ dinv, int n) {
    int i = blockIdx.x * blockDim.x + threadIdx.x;
    if (i < n) { float v = deg[i]; dinv[i] = v > 0.f ? rsqrtf(v) : 0.f; }
}

// f[n,:] = bias + dinv[n]^2 * h[n,:]   (self-loop term + bias init)
__global__ void k_gcn_init(const float* __restrict__ h, const float* __restrict__ dinv,
                           const float* __restrict__ bias, float* __restrict__ f, int ldf)
{
    int n = blockIdx.x, d = threadIdx.x;
    float di = dinv[n];
    f[(size_t)n * ldf + d] = bias[d] + di * di * h[(size_t)n * D128 + d];
}

// f[dst,:] += dinv[src]*ew*dinv[dst] * h[src,:]
__global__ void k_gcn_scatter(const int* __restrict__ src, const int* __restrict__ dst,
                              const float* __restrict__ ew, const float* __restrict__ dinv,
                              const float* __restrict__ h, float* __restrict__ f, int ldf)
{
    int e = blockIdx.x, d = threadIdx.x;
    int s = src[e], t = dst[e];
    float norm = dinv[s] * ew[e] * dinv[t];
    atomicAdd(&f[(size_t)t * ldf + d], norm * h[(size_t)s * D128 + d]);
}

__global__ void k_relu_ld(float* __restrict__ f, int ldf) {
    int n = blockIdx.x, d = threadIdx.x;
    float v = f[(size_t)n * ldf + d];
    f[(size_t)n * ldf + d] = v > 0.f ? v : 0.f;
}

// per-(node,head) attention dot products:  an_src[n,h] = <hg[n,h,:], a_src[h,:]>, same for dst
__global__ void k_att_node(const float* __restrict__ hg, const float* __restrict__ wsrc,
                           const float* __restrict__ wdst, float* __restrict__ an_src,
                           float* __restrict__ an_dst, int Nn)
{
    int i = blockIdx.x * blockDim.x + threadIdx.x;
    if (i >= Nn * H8) return;
    int n = i >> 3, h = i & 7;
    const float* hp = hg + (size_t)n * (H8 * D128) + h * D128;
    const float* ws = wsrc + h * D128;
    const float* wd = wdst + h * D128;
    float s0 = 0.f, s1 = 0.f;
    for (int d = 0; d < D128; ++d) { float v = hp[d]; s0 += v * ws[d]; s1 += v * wd[d]; }
    an_src[i] = s0; an_dst[i] = s1;
}

__global__ void k_sum(const float* __restrict__ x, int n, float* out) {
    __shared__ float sm[256];
    int i = blockIdx.x * blockDim.x + threadIdx.x;
    sm[threadIdx.x] = (i < n) ? x[i] : 0.f;
    __syncthreads();
    for (int o = 128; o > 0; o >>= 1) {
        if ((int)threadIdx.x < o) sm[threadIdx.x] += sm[threadIdx.x + o];
        __syncthreads();
    }
    if (threadIdx.x == 0) atomicAdd(out, sm[0]);
}

// c_edge[h] = sum_d We[0, h*D+d] * a_edge[h,d]
__global__ void k_cedge(const float* __restrict__ we, const float* __restrict__ aedge,
                        float* __restrict__ cedge)
{
    int h = threadIdx.x; // 8 threads
    float s = 0.f;
    for (int d = 0; d < D128; ++d) s += we[h * D128 + d] * aedge[h * D128 + d];
    cedge[h] = s;
}

// ordered-uint mapping for float atomicMax
__device__ __forceinline__ unsigned f2s(float f) {
    unsigned u = __float_as_uint(f);
    return (u & 0x80000000u) ? ~u : (u | 0x80000000u);
}
__device__ __forceinline__ float s2f(unsigned s) {
    return (s & 0x80000000u) ? __uint_as_float(s & 0x7FFFFFFFu) : __uint_as_float(~s);
}

// attention logits + leaky_relu + segment max (self-loops appended at i>=E with mean edge attr)
__global__ void k_alpha(const int* __restrict__ src, const int* __restrict__ dst,
                        const float* __restrict__ ew, const float* __restrict__ easum,
                        const float* __restrict__ an_src, const float* __restrict__ an_dst,
                        const float* __restrict__ cedge, float* __restrict__ alpha,
                        unsigned* __restrict__ amax, int E, int Nn)
{
    int i = blockIdx.x * blockDim.x + threadIdx.x;
    if (i >= E + Nn) return;
    int s, t; float w;
    if (i < E) { s = src[i]; t = dst[i]; w = ew[i]; }
    else       { s = t = i - E;          w = easum[0] / (float)E; }
    for (int h = 0; h < H8; ++h) {
        float a = an_src[s * H8 + h] + an_dst[t * H8 + h] + w * cedge[h];
        a = a > 0.f ? a : 0.2f * a;
        alpha[(size_t)i * H8 + h] = a;
        atomicMax(&amax[t * H8 + h], f2s(a));
    }
}

// ex = exp(alpha - max); den[dst,h] += ex  (alpha overwritten with ex)
__global__ void k_expden(const int* __restrict__ dst, float* __restrict__ alpha,
                         const unsigned* __restrict__ amax, float* __restrict__ den,
                         int E, int Nn)
{
    int i = blockIdx.x * blockDim.x + threadIdx.x;
    if (i >= E + Nn) return;
    int t = (i < E) ? dst[i] : (i - E);
    for (int h = 0; h < H8; ++h) {
        float ex = expf(alpha[(size_t)i * H8 + h] - s2f(amax[t * H8 + h]));
        alpha[(size_t)i * H8 + h] = ex;
        atomicAdd(&den[t * H8 + h], ex);
    }
}

// acc[dst,d] += (1/H) * sum_h (ex/(den+eps)) * hg[src,h,d]   — block per edge, 128 lanes over d
__global__ void k_gat_scatter(const int* __restrict__ src, const int* __restrict__ dst,
                              const float* __restrict__ ex, const float* __restrict__ den,
                              const float* __restrict__ hg, float* __restrict__ acc, int E)
{
    int i = blockIdx.x, d = threadIdx.x;
    int s, t;
    if (i < E) { s = src[i]; t = dst[i]; } else { s = t = i - E; }
    float sum = 0.f;
    for (int h = 0; h < H8; ++h) {
        float coef = ex[(size_t)i * H8 + h] / (den[t * H8 + h] + 1e-16f);
        sum += coef * hg[(size_t)s * (H8 * D128) + h * D128 + d];
    }
    atomicAdd(&acc[(size_t)t * D128 + d], sum * (1.0f / H8));
}

__global__ void k_bias_relu(const float* __restrict__ acc, const float* __restrict__ bias,
                            float* __restrict__ att)
{
    int n = blockIdx.x, d = threadIdx.x;
    float v = acc[(size_t)n * D128 + d] + bias[d];
    att[(size_t)n * D128 + d] = v > 0.f ? v : 0.f;
}

__global__ void k_transpose(const float* __restrict__ A, float* __restrict__ AT, int R, int Cc) {
    int i = blockIdx.x * blockDim.x + threadIdx.x;
    if (i >= R * Cc) return;
    int r = i / Cc, c = i % Cc;
    AT[(size_t)c * R + r] = A[i];
}

// ================= host-side orchestration =================
static inline int ceil_div(int a, int b) { return (a + b - 1) / b; }

struct BranchWS {
    float *ew, *deg, *dinv, *h, *hg, *an_src, *an_dst, *cedge, *easum, *alpha, *den, *acc, *att, *fcat, *wT;
    unsigned* amax;
};

static void run_branch(const float* x, const int* edges, const float* mat,
                       const float* gw1, const float* gb1,
                       const float* aw, const float* a_src, const float* a_dst, const float* a_edge,
                       const float* awe, const float* ab,
                       const float* gw2, const float* gb2,
                       const float* convw, const float* convb,
                       int Nn, int E, float* fea, const BranchWS& w, hipStream_t st)
{
    const int* src = edges;
    const int* dst = edges + E;
    const int EN = E + Nn;

    k_fillf<<<ceil_div(Nn, 256), 256, 0, st>>>(w.deg, 1.0f, Nn);
    k_edge_w_deg<<<ceil_div(E, 256), 256, 0, st>>>(src, dst, mat, Nn, w.ew, w.deg, E);
    k_dinv<<<ceil_div(Nn, 256), 256, 0, st>>>(w.deg, w.dinv, Nn);

    // GCN1: h = x @ gw1 ; f1 = relu(bias + norm-scatter)
    k_transpose<<<ceil_div(D128 * D128, 256), 256, 0, st>>>(gw1, w.wT, D128, D128);
    k_gemm_wmma<<<dim3(ceil_div(D128, 32), ceil_div(Nn, 16)), 32, 0, st>>>(
        x, D128, w.wT, D128, w.h, D128, nullptr, Nn, D128, D128);
    k_gcn_init<<<Nn, D128, 0, st>>>(w.h, w.dinv, gb1, w.fcat, 256);
    k_gcn_scatter<<<E, D128, 0, st>>>(src, dst, w.ew, w.dinv, w.h, w.fcat, 256);
    k_relu_ld<<<Nn, D128, 0, st>>>(w.fcat, 256);

    // GAT: hg = f1 @ aw  [N, 1024]
    k_transpose<<<ceil_div(D128 * H8 * D128, 256), 256, 0, st>>>(aw, w.wT, D128, H8 * D128);
    k_gemm_wmma<<<dim3(ceil_div(H8 * D128, 32), ceil_div(Nn, 16)), 32, 0, st>>>(
        w.fcat, 256, w.wT, D128, w.hg, H8 * D128, nullptr, Nn, H8 * D128, D128);
    k_att_node<<<ceil_div(Nn * H8, 256), 256, 0, st>>>(w.hg, a_src, a_dst, w.an_src, w.an_dst, Nn);
    k_fillf<<<1, 1, 0, st>>>(w.easum, 0.f, 1);
    k_sum<<<ceil_div(E, 256), 256, 0, st>>>(w.ew, E, w.easum);
    k_cedge<<<1, H8, 0, st>>>(awe, a_edge, w.cedge);
    k_fillu<<<ceil_div(Nn * H8, 256), 256, 0, st>>>(w.amax, 0u, Nn * H8);
    k_alpha<<<ceil_div(EN, 128), 128, 0, st>>>(src, dst, w.ew, w.easum, w.an_src, w.an_dst,
                                               w.cedge, w.alpha, w.amax, E, Nn);
    k_fillf<<<ceil_div(Nn * H8, 256), 256, 0, st>>>(w.den, 0.f, Nn * H8);
    k_expden<<<ceil_div(EN, 128), 128, 0, st>>>(dst, w.alpha, w.amax, w.den, E, Nn);
    k_fillf<<<ceil_div(Nn * D128, 256), 256, 0, st>>>(w.acc, 0.f, Nn * D128);
    k_gat_scatter<<<EN, D128, 0, st>>>(src, dst, w.alpha, w.den, w.hg, w.acc, E);
    k_bias_relu<<<Nn, D128, 0, st>>>(w.acc, ab, w.att);

    // GCN2: h = att @ gw2 ; f2 = relu(bias + norm-scatter)  (written at fcat+128, ld 256)
    k_transpose<<<ceil_div(D128 * D128, 256), 256, 0, st>>>(gw2, w.wT, D128, D128);
    k_gemm_wmma<<<dim3(ceil_div(D128, 32), ceil_div(Nn, 16)), 32, 0, st>>>(
        w.att, D128, w.wT, D128, w.h, D128, nullptr, Nn, D128, D128);
    k_gcn_init<<<Nn, D128, 0, st>>>(w.h, w.dinv, gb2, w.fcat + D128, 256);
    k_gcn_scatter<<<E, D128, 0, st>>>(src, dst, w.ew, w.dinv, w.h, w.fcat + D128, 256);
    k_relu_ld<<<Nn, D128, 0, st>>>(w.fcat + D128, 256);

    // CNN: fea[N,256] = [f1|f2] @ Wcat^T + b.  Wcat[o,k] = convw[o*256+k] is ALREADY
    // the transposed-B layout ([N=256 cols][K=256]) the GEMM wants -> no transpose.
    k_gemm_wmma<<<dim3(ceil_div(256, 32), ceil_div(Nn, 16)), 32, 0, st>>>(
        w.fcat, 256, convw, 256, fea, 256, convb, Nn, 256, 256);
}

extern "C" void kernel_launch(void* const* d_in, const int* in_sizes, int n_in,
                              void* d_out, int out_size, void* d_ws, size_t ws_size,
                              hipStream_t stream)
{
    (void)in_sizes; (void)n_in; (void)out_size; (void)ws_size;
    const float* x_cir     = (const float*)d_in[0];
    const float* x_dis     = (const float*)d_in[1];
    const float* cc_mat    = (const float*)d_in[2];
    const float* dd_mat    = (const float*)d_in[3];
    const float* gcn_c1_w  = (const float*)d_in[4];
    const float* gcn_c1_b  = (const float*)d_in[5];
    const float* gat_c_w   = (const float*)d_in[6];
    const float* gat_c_as  = (const float*)d_in[7];
    const float* gat_c_ad  = (const float*)d_in[8];
    const float* gat_c_ae  = (const float*)d_in[9];
    const float* gat_c_we  = (const float*)d_in[10];
    const float* gat_c_b   = (const float*)d_in[11];
    const float* gcn_c2_w  = (const float*)d_in[12];
    const float* gcn_c2_b  = (const float*)d_in[13];
    const float* gcn_d1_w  = (const float*)d_in[14];
    const float* gcn_d1_b  = (const float*)d_in[15];
    const float* gat_d_w   = (const float*)d_in[16];
    const float* gat_d_as  = (const float*)d_in[17];
    const float* gat_d_ad  = (const float*)d_in[18];
    const float* gat_d_ae  = (const float*)d_in[19];
    const float* gat_d_we  = (const float*)d_in[20];
    const float* gat_d_b   = (const float*)d_in[21];
    const float* gcn_d2_w  = (const float*)d_in[22];
    const float* gcn_d2_b  = (const float*)d_in[23];
    const float* conv_c_w  = (const float*)d_in[24];
    const float* conv_c_b  = (const float*)d_in[25];
    const float* conv_d_w  = (const float*)d_in[26];
    const float* conv_d_b  = (const float*)d_in[27];
    const int*   cc_edges  = (const int*)d_in[28];
    const int*   dd_edges  = (const int*)d_in[29];

    float* out     = (float*)d_out;
    float* scores  = out;                          // [585, 88]
    float* cir_fea = out + 585 * 88;               // [585, 256]
    float* dis_fea = out + 585 * 88 + 585 * 256;   // [88, 256]

    // workspace carve (sized for the larger branch, reused by both)
    float* ws = (float*)d_ws;
    size_t off = 0;
    auto take = [&](size_t n) -> float* {
        float* p = ws + off;
        off += (n + 3) & ~(size_t)3;
        return p;
    };
    const int NMAX = 585, EMAX = 20000;
    BranchWS w;
    w.ew     = take(EMAX);
    w.deg    = take(NMAX);
    w.dinv   = take(NMAX);
    w.h      = take((size_t)NMAX * D128);
    w.hg     = take((size_t)NMAX * H8 * D128);
    w.an_src = take((size_t)NMAX * H8);
    w.an_dst = take((size_t)NMAX * H8);
    w.cedge  = take(H8);
    w.easum  = take(1);
    w.alpha  = take((size_t)(EMAX + NMAX) * H8);
    w.amax   = (unsigned*)take((size_t)NMAX * H8);
    w.den    = take((size_t)NMAX * H8);
    w.acc    = take((size_t)NMAX * D128);
    w.att    = take((size_t)NMAX * D128);
    w.fcat   = take((size_t)NMAX * 256);
    w.wT     = take((size_t)H8 * D128 * D128);     // shared transposed-weight buffer (max 1024x128)

    run_branch(x_cir, cc_edges, cc_mat, gcn_c1_w, gcn_c1_b, gat_c_w, gat_c_as, gat_c_ad, gat_c_ae,
               gat_c_we, gat_c_b, gcn_c2_w, gcn_c2_b, conv_c_w, conv_c_b,
               585, 20000, cir_fea, w, stream);
    run_branch(x_dis, dd_edges, dd_mat, gcn_d1_w, gcn_d1_b, gat_d_w, gat_d_as, gat_d_ad, gat_d_ae,
               gat_d_we, gat_d_b, gcn_d2_w, gcn_d2_b, conv_d_w, conv_d_b,
               88, 3000, dis_fea, w, stream);

    // scores = cir_fea @ dis_fea^T : dis_fea [88,256] is already the Bt layout -> no transpose
    k_gemm_wmma<<<dim3(ceil_div(88, 32), ceil_div(585, 16)), 32, 0, stream>>>(
        cir_fea, 256, dis_fea, 256, scores, 88, nullptr, 585, 88, 256);
}